// SO3SE3Conv_48679159332867
// MI455X (gfx1250) — compile-verified
//
#include <hip/hip_runtime.h>
#include <hip/hip_bf16.h>

typedef float v2f __attribute__((ext_vector_type(2)));
typedef float v8f __attribute__((ext_vector_type(8)));

#define VV    642
#define S3    1728
#define NCOL  13824     // Fin * S3
#define FIN   8
#define KCH   5
#define NCH   40        // Fin * K
#define FOUT  16
#define LA_STRIDE 644   // 16-row L tile, K padded 642->644 (zeros), odd-ish bank stride
#define LW_STRIDE 1081  // 1080 + 1 pad: odd stride -> conflict-free lane reads

// ---------------------------------------------------------------------------
// Fused weight build: Wf[fo][c][t], c = k*8 + fin (k-major), t = dz*9+dy*3+dx
// ---------------------------------------------------------------------------
__global__ __launch_bounds__(256)
void build_weight_kernel(const float* __restrict__ wSph,   // (16,8,5)
                         const float* __restrict__ wSpa,   // (16,8,1,1,1,4)
                         float* __restrict__ Wf)           // (16,40,27)
{
    int idx = blockIdx.x * blockDim.x + threadIdx.x;
    if (idx >= FOUT * NCH * 27) return;
    int t  = idx % 27;
    int c  = (idx / 27) % NCH;
    int fo = idx / (27 * NCH);
    int k   = c >> 3;
    int fin = c & 7;
    int dx = t % 3 - 1, dy = (t / 3) % 3 - 1, dz = t / 9 - 1;
    int d2 = dx * dx + dy * dy + dz * dz;        // distance-class index
    Wf[idx] = wSph[(fo * FIN + fin) * KCH + k] * wSpa[(fo * FIN + fin) * 4 + d2];
}

// ---------------------------------------------------------------------------
// Chebyshev GEMM step: Xout = alpha * (L @ Bsrc) + beta * prev2
// A-tile (16 x 644, zero-padded) staged in LDS, shared by the 4 waves/block.
// K-loop is fully branchless: A pad zeros annihilate clamped B tail rows.
// ---------------------------------------------------------------------------
template <bool BIN, bool P2IN, bool HASBETA>
__global__ __launch_bounds__(128)
void cheb_gemm_kernel(const float* __restrict__ L,
                      const float* __restrict__ Bsrc,
                      const float* __restrict__ prev2,
                      float* __restrict__ Xout,
                      float alpha, float beta)
{
    const int NT = NCOL / 64;                       // 216 column groups
    __shared__ __align__(16) float lA[16 * LA_STRIDE];

    int tid  = threadIdx.x;
    int wid  = blockIdx.x * 4 + (tid >> 5);
    int lane = tid & 31;
    int half = lane >> 4, l16 = lane & 15;
    int mt   = wid / NT;                            // block-uniform (216 % 4 == 0)
    int ng   = wid % NT;

    // ---- stage L tile into LDS, zero-padded (rows >= VV, cols >= VV -> 0) ----
    {
        int base = mt * 16;
        for (int i = tid; i < 16 * LA_STRIDE; i += 128) {
            int r  = i / LA_STRIDE, cc = i - r * LA_STRIDE;
            int gr = base + r;
            int grc = gr < VV ? gr : VV - 1;
            int ccc = cc < VV ? cc : VV - 1;
            float val = L[grc * VV + ccc];
            lA[i] = (gr < VV && cc < VV) ? val : 0.f;
        }
    }
    __syncthreads();

    int cb0 = ng * 64;
    int  col[4];
    int  boffIn[4];
    #pragma unroll
    for (int j = 0; j < 4; ++j) {
        col[j] = cb0 + 16 * j + l16;
        int fin = (cb0 + 16 * j) / S3;              // constant over 16 cols
        boffIn[j] = fin * (VV - 1) * S3;            // < 2^31
    }

    v8f acc[4] = {};
    const float* lArow = &lA[l16 * LA_STRIDE];

    for (int k0 = 0; k0 < 644; k0 += 4) {
        int kk = k0 + 2 * half;                     // K = 2*half + {0,1}
        v2f a = *reinterpret_cast<const v2f*>(lArow + kk);   // ds_load_b64
        int r0 = kk     < VV ? kk     : VV - 1;     // clamped B rows (A pad = 0)
        int r1 = kk + 1 < VV ? kk + 1 : VV - 1;
        #pragma unroll
        for (int j = 0; j < 4; ++j) {
            v2f b;
            if (BIN) {
                b.x = Bsrc[r0 * S3 + col[j] + boffIn[j]];
                b.y = Bsrc[r1 * S3 + col[j] + boffIn[j]];
            } else {
                b.x = Bsrc[(long)r0 * NCOL + col[j]];
                b.y = Bsrc[(long)r1 * NCOL + col[j]];
            }
            acc[j] = __builtin_amdgcn_wmma_f32_16x16x4_f32(
                false, a, false, b, (short)0, acc[j], false, false);
        }
    }

    // ---- epilogue: recurrence combine + store (row-guarded, last tile only) ----
    #pragma unroll
    for (int g = 0; g < 8; ++g) {
        int r = mt * 16 + g + 8 * half;
        if (r < VV) {
            #pragma unroll
            for (int j = 0; j < 4; ++j) {
                float vo = alpha * acc[j][g];
                if (HASBETA) {
                    float p = P2IN ? prev2[r * S3 + col[j] + boffIn[j]]
                                   : prev2[(long)r * NCOL + col[j]];
                    vo += beta * p;
                }
                Xout[(long)r * NCOL + col[j]] = vo;
            }
        }
    }
}

// ---------------------------------------------------------------------------
// Conv stage: per vertex v, Out(16,1728) = sum_t Wf_t(16,40) @ Shift_t(40,1728).
// Fused weight staged in LDS; SAME-pad masks hoisted per tap and applied as
// branchless 0/1 multiplies on clamp-addressed unconditional loads.
// ---------------------------------------------------------------------------
__global__ __launch_bounds__(128)
void conv_kernel(const float* __restrict__ inputs,
                 const float* __restrict__ x1, const float* __restrict__ x2,
                 const float* __restrict__ x3, const float* __restrict__ x4,
                 const float* __restrict__ Wf,
                 const float* __restrict__ bias,
                 float* __restrict__ out)
{
    const int NG = S3 / 64;                         // 27 column groups
    __shared__ __align__(16) float lW[FOUT * LW_STRIDE];

    int tid  = threadIdx.x;
    // ---- stage fused weight into LDS (padded stride for conflict-free reads) ----
    for (int i = tid; i < FOUT * NCH * 27; i += 128) {
        int fo = i / (NCH * 27), rest = i - fo * (NCH * 27);
        lW[fo * LW_STRIDE + rest] = Wf[i];
    }
    __syncthreads();

    int wid  = blockIdx.x * 4 + (tid >> 5);
    int lane = tid & 31;
    int half = lane >> 4, l16 = lane & 15;
    int v  = wid / NG;
    int ng = wid % NG;
    if (v >= VV) return;

    const float* kbuf[KCH] = { inputs, x1, x2, x3, x4 };
    const float* lWrow = &lW[l16 * LW_STRIDE];

    int scol[4], px[4], py[4], pz[4];
    #pragma unroll
    for (int j = 0; j < 4; ++j) {
        int s = ng * 64 + 16 * j + l16;
        scol[j] = s;
        px[j] = s % 12;
        py[j] = (s / 12) % 12;
        pz[j] = s / 144;
    }

    v8f acc[4] = {};

    for (int t = 0; t < 27; ++t) {
        int dx = t % 3 - 1, dy = (t / 3) % 3 - 1, dz = t / 9 - 1;
        int shift = dz * 144 + dy * 12 + dx;

        // hoisted per-tap masks + clamped in-slice offsets (always legal addresses)
        float m[4]; int spc[4];
        #pragma unroll
        for (int j = 0; j < 4; ++j) {
            bool ok = (unsigned)(px[j] + dx) < 12u &&
                      (unsigned)(py[j] + dy) < 12u &&
                      (unsigned)(pz[j] + dz) < 12u;
            m[j] = ok ? 1.f : 0.f;
            int sp = scol[j] + shift;
            sp = sp < 0 ? 0 : (sp > S3 - 1 ? S3 - 1 : sp);
            spc[j] = sp;
        }

        for (int k0 = 0; k0 < NCH; k0 += 4) {
            int c   = k0 + 2 * half;                // channels {c, c+1}
            int k   = k0 >> 3;                      // wave-uniform buffer id
            int fin = c & 7;
            const float* buf = kbuf[k];

            v2f a;
            a.x = lWrow[c * 27 + t];                // ds_load
            a.y = lWrow[(c + 1) * 27 + t];

            #pragma unroll
            for (int j = 0; j < 4; ++j) {
                v2f b;
                if (k == 0) {                       // x0 straight from inputs
                    long o = (long)fin * VV * S3 + (long)v * S3 + spc[j];
                    b.x = buf[o] * m[j];
                    b.y = buf[o + (long)VV * S3] * m[j];
                } else {                            // x_k workspace [v][fin*S3+s]
                    long o = (long)v * NCOL + fin * S3 + spc[j];
                    b.x = buf[o] * m[j];
                    b.y = buf[o + S3] * m[j];
                }
                acc[j] = __builtin_amdgcn_wmma_f32_16x16x4_f32(
                    false, a, false, b, (short)0, acc[j], false, false);
            }
        }
    }

    // ---- store: out[fo][v][s] + bias, fo = g + 8*half (all rows valid) ----
    #pragma unroll
    for (int g = 0; g < 8; ++g) {
        int fo = g + 8 * half;
        float bi = bias[fo];
        #pragma unroll
        for (int j = 0; j < 4; ++j)
            out[(long)fo * VV * S3 + (long)v * S3 + scol[j]] = acc[j][g] + bi;
    }
}

// ---------------------------------------------------------------------------
extern "C" void kernel_launch(void* const* d_in, const int* in_sizes, int n_in,
                              void* d_out, int out_size, void* d_ws, size_t ws_size,
                              hipStream_t stream) {
    const float* lap   = (const float*)d_in[0];   // (642,642)
    const float* inp   = (const float*)d_in[1];   // (1,8,642,12,12,12)
    const float* wSph  = (const float*)d_in[2];   // (16,8,5)
    const float* wSpa  = (const float*)d_in[3];   // (16,8,1,1,1,4)
    const float* bias  = (const float*)d_in[4];   // (16,)
    float* out = (float*)d_out;

    float* Wf = (float*)d_ws;
    const size_t wfPad  = ((size_t)FOUT * NCH * 27 * sizeof(float) + 511) & ~(size_t)511;
    const size_t xElems = (size_t)VV * NCOL;
    float* xb = (float*)((char*)d_ws + wfPad);
    float* x1 = xb;
    float* x2 = xb + xElems;
    float* x3 = xb + 2 * xElems;
    float* x4 = xb + 3 * xElems;

    build_weight_kernel<<<(FOUT * NCH * 27 + 255) / 256, 256, 0, stream>>>(wSph, wSpa, Wf);

    const int gemmBlocks = (41 * (NCOL / 64)) / 4;       // 8856 waves / 4
    cheb_gemm_kernel<true,  true,  false><<<gemmBlocks, 128, 0, stream>>>(lap, inp, inp, x1, 1.f,  0.f);
    cheb_gemm_kernel<false, true,  true ><<<gemmBlocks, 128, 0, stream>>>(lap, x1,  inp, x2, 2.f, -1.f);
    cheb_gemm_kernel<false, false, true ><<<gemmBlocks, 128, 0, stream>>>(lap, x2,  x1,  x3, 2.f, -1.f);
    cheb_gemm_kernel<false, false, true ><<<gemmBlocks, 128, 0, stream>>>(lap, x3,  x2,  x4, 2.f, -1.f);

    const int convWaves  = VV * (S3 / 64);               // 17334
    const int convBlocks = (convWaves + 3) / 4;
    conv_kernel<<<convBlocks, 128, 0, stream>>>(inp, x1, x2, x3, x4, Wf, bias, out);
}